// EETLoss_24721831755951
// MI455X (gfx1250) — compile-verified
//
#include <hip/hip_runtime.h>
#include <hip/hip_bf16.h>

typedef __attribute__((ext_vector_type(2))) float v2f;
typedef __attribute__((ext_vector_type(8))) float v8f;

#define NROWS 4096
#define DIM   1024
#define LMARGIN 0.3f
#define LEPS 1e-12f

#define KC 64                    // k-chunk staged in LDS
#define CG 128                   // columns per group = 8 waves * 16
#define PITCH 68                 // LDS pitch (floats): 8B-aligned rows, no bank conflicts
#define CHUNKF ((16 + CG) * PITCH)   // floats per staging buffer (A rows + B rows)

// ---------------- row sum-of-squares: one wave per row ----------------
__global__ void __launch_bounds__(256)
k_rowsumsq(const float* __restrict__ X, float* __restrict__ sq) {
    int wave = (blockIdx.x * blockDim.x + threadIdx.x) >> 5;
    int lane = threadIdx.x & 31;
    if (wave >= NROWS) return;
    const float* row = X + (size_t)wave * DIM;
    float s = 0.f;
    #pragma unroll
    for (int k = lane * 4; k < DIM; k += 32 * 4) {
        float4 v = *(const float4*)(row + k);
        s += v.x * v.x + v.y * v.y + v.z * v.z + v.w * v.w;
    }
    #pragma unroll
    for (int off = 16; off; off >>= 1) s += __shfl_xor(s, off, 32);
    if (lane == 0) sq[wave] = s;
}

// ------- fused GEMM (fp32 WMMA) + masked hardest-pos/neg mining -------
// one workgroup (8 waves) per 16-row tile; async double-buffered staging.
__global__ void __launch_bounds__(256)
k_hardmine(const float* __restrict__ X, const int* __restrict__ labels,
           const float* __restrict__ sq,
           float* __restrict__ d_ap, float* __restrict__ d_an,
           int* __restrict__ ind_p, int* __restrict__ ind_n) {
    __shared__ float lds[2 * CHUNKF];     // double-buffered A(16xKC) + B(CGxKC)
    __shared__ float sqA[16];
    __shared__ int   labA[16];
    __shared__ float red_pv[8][16];
    __shared__ int   red_pi[8][16];
    __shared__ float red_nv[8][16];
    __shared__ int   red_ni[8][16];

    const int tid  = threadIdx.x;
    const int wave = tid >> 5;
    const int lane = tid & 31;
    const int row0 = blockIdx.x * 16;

    if (tid < 16) { sqA[tid] = sq[row0 + tid]; labA[tid] = labels[row0 + tid]; }

    const unsigned ldsbase = (unsigned)(uintptr_t)(void*)lds;  // LDS byte offset
    const int ln16  = lane & 15;
    const int khalf = (lane >> 4) << 1;   // 0 (lanes 0-15) or 2 (lanes 16-31)
    const int mbase = (lane >> 4) << 3;   // rows 0-7 or 8-15 per C/D layout

    // issue one chunk = 72 async B128 transfers (9 per wave), direct global->LDS
    auto issue = [&](int col0n, int kcn, int bufb /*float offset of dest buffer*/) {
        #pragma unroll
        for (int i = 0; i < 9; ++i) {
            int g = ((wave * 9 + i) << 5) + lane;   // 0..2303, A/B split at 256 (wave-uniform)
            int ldsrow, grow;
            if (g < 256) { int r = g >> 4;        ldsrow = r;      grow = row0  + r; }
            else         { int r = (g - 256) >> 4; ldsrow = 16 + r; grow = col0n + r; }
            int c = (g & 15) << 2;                // float4 column within chunk
            unsigned ldsb = ldsbase + (unsigned)((bufb + ldsrow * PITCH + c) << 2);
            unsigned goff = (unsigned)(((grow << 10) + kcn + c) << 2);
            asm volatile("global_load_async_to_lds_b128 %0, %1, %2"
                         :: "v"(ldsb), "v"(goff), "s"(X) : "memory");
        }
    };

    float pv[8]; int pi[8]; float nv[8]; int ni[8];
    #pragma unroll
    for (int r = 0; r < 8; ++r) {
        pv[r] = -__builtin_inff(); pi[r] = 0x7fffffff;
        nv[r] =  __builtin_inff(); ni[r] = 0x7fffffff;
    }

    issue(0, 0, 0);                       // prefetch chunk (cg=0, kc=0) into buffer 0
    int parity = 0;

    for (int cg = 0; cg < NROWS / CG; ++cg) {
        const int col0 = cg * CG;
        v8f acc0 = {}, acc1 = {};         // two independent WMMA chains
        for (int kc = 0; kc < DIM; kc += KC) {
            int kcn = kc + KC, cgn = cg;
            if (kcn == DIM) { kcn = 0; ++cgn; }
            if (cgn < NROWS / CG) {       // uniform branch
                issue(cgn * CG, kcn, parity ? 0 : CHUNKF);
                asm volatile("s_wait_asynccnt 0x9" ::: "memory"); // cur chunk done
            } else {
                asm volatile("s_wait_asynccnt 0x0" ::: "memory");
            }
            __syncthreads();              // all waves' async data visible in LDS
            const float* Ar = lds + parity * CHUNKF + ln16 * PITCH + khalf;
            const float* Br = lds + parity * CHUNKF + (16 + wave * 16 + ln16) * PITCH + khalf;
            #pragma unroll
            for (int k = 0; k < KC; k += 8) {
                v2f a0, b0, a1, b1;
                a0.x = Ar[k];     a0.y = Ar[k + 1];
                b0.x = Br[k];     b0.y = Br[k + 1];
                a1.x = Ar[k + 4]; a1.y = Ar[k + 5];
                b1.x = Br[k + 4]; b1.y = Br[k + 5];
                acc0 = __builtin_amdgcn_wmma_f32_16x16x4_f32(
                           false, a0, false, b0, (short)0, acc0, false, false);
                acc1 = __builtin_amdgcn_wmma_f32_16x16x4_f32(
                           false, a1, false, b1, (short)0, acc1, false, false);
            }
            __syncthreads();              // buffer free before it is re-issued
            parity ^= 1;
        }
        // epilogue: dot -> distance, masked running max/min with first-index ties
        const int j    = col0 + wave * 16 + ln16;
        const float sj = sq[j];
        const int  lj  = labels[j];
        #pragma unroll
        for (int r = 0; r < 8; ++r) {
            const int m = mbase + r;
            float dot  = acc0[r] + acc1[r];
            float dist = sqrtf(fmaxf(sqA[m] + sj - 2.0f * dot, LEPS));
            if (labA[m] == lj) {
                if (dist > pv[r] || (dist == pv[r] && j < pi[r])) { pv[r] = dist; pi[r] = j; }
            } else {
                if (dist < nv[r] || (dist == nv[r] && j < ni[r])) { nv[r] = dist; ni[r] = j; }
            }
        }
    }

    // cross-lane combine within each 16-lane half (masks < 16 keep halves apart)
    #pragma unroll
    for (int r = 0; r < 8; ++r) {
        #pragma unroll
        for (int m = 8; m >= 1; m >>= 1) {
            float opv = __shfl_xor(pv[r], m, 32); int opi = __shfl_xor(pi[r], m, 32);
            float onv = __shfl_xor(nv[r], m, 32); int oni = __shfl_xor(ni[r], m, 32);
            if (opv > pv[r] || (opv == pv[r] && opi < pi[r])) { pv[r] = opv; pi[r] = opi; }
            if (onv < nv[r] || (onv == nv[r] && oni < ni[r])) { nv[r] = onv; ni[r] = oni; }
        }
    }
    if (ln16 == 0) {  // lane 0 -> rows 0-7, lane 16 -> rows 8-15
        #pragma unroll
        for (int r = 0; r < 8; ++r) {
            red_pv[wave][mbase + r] = pv[r]; red_pi[wave][mbase + r] = pi[r];
            red_nv[wave][mbase + r] = nv[r]; red_ni[wave][mbase + r] = ni[r];
        }
    }
    __syncthreads();
    if (tid < 16) {
        float bpv = -__builtin_inff(); int bpi = 0x7fffffff;
        float bnv =  __builtin_inff(); int bni = 0x7fffffff;
        for (int w = 0; w < 8; ++w) {
            float cpv = red_pv[w][tid]; int cpi = red_pi[w][tid];
            float cnv = red_nv[w][tid]; int cni = red_ni[w][tid];
            if (cpv > bpv || (cpv == bpv && cpi < bpi)) { bpv = cpv; bpi = cpi; }
            if (cnv < bnv || (cnv == bnv && cni < bni)) { bnv = cnv; bni = cni; }
        }
        d_ap[row0 + tid] = bpv; ind_p[row0 + tid] = bpi;
        d_an[row0 + tid] = bnv; ind_n[row0 + tid] = bni;
    }
}

// -------- d_pn[i] = dist(ind_p[i], ind_n[i]) : one wave per row --------
__global__ void __launch_bounds__(256)
k_dpn(const float* __restrict__ X, const float* __restrict__ sq,
      const int* __restrict__ ind_p, const int* __restrict__ ind_n,
      float* __restrict__ d_pn) {
    int i = (blockIdx.x * blockDim.x + threadIdx.x) >> 5;
    int lane = threadIdx.x & 31;
    if (i >= NROWS) return;
    int p = ind_p[i], n = ind_n[i];
    const float* xp = X + (size_t)p * DIM;
    const float* xn = X + (size_t)n * DIM;
    float s = 0.f;
    #pragma unroll
    for (int k = lane * 4; k < DIM; k += 32 * 4) {
        float4 a = *(const float4*)(xp + k);
        float4 b = *(const float4*)(xn + k);
        s += a.x * b.x + a.y * b.y + a.z * b.z + a.w * b.w;
    }
    #pragma unroll
    for (int off = 16; off; off >>= 1) s += __shfl_xor(s, off, 32);
    if (lane == 0) d_pn[i] = sqrtf(fmaxf(sq[p] + sq[n] - 2.0f * s, LEPS));
}

// ---------------- final scalar loss (all terms share 1/n) ----------------
__global__ void __launch_bounds__(256)
k_loss(const float* __restrict__ d_ap, const float* __restrict__ d_an,
       const float* __restrict__ d_pn, float* __restrict__ out) {
    __shared__ float sdata[256];
    float s = 0.f;
    for (int i = threadIdx.x; i < NROWS; i += 256) {
        float ap = d_ap[i], an = d_an[i], pn = d_pn[i];
        s += fmaxf(0.f, ap + LMARGIN - an)
           + fmaxf(0.f, ap + LMARGIN - pn)
           + fabsf(pn - an);
    }
    sdata[threadIdx.x] = s;
    __syncthreads();
    for (int off = 128; off; off >>= 1) {
        if (threadIdx.x < off) sdata[threadIdx.x] += sdata[threadIdx.x + off];
        __syncthreads();
    }
    if (threadIdx.x == 0) out[0] = sdata[0] / (float)NROWS;
}

extern "C" void kernel_launch(void* const* d_in, const int* in_sizes, int n_in,
                              void* d_out, int out_size, void* d_ws, size_t ws_size,
                              hipStream_t stream) {
    const float* X      = (const float*)d_in[0];
    const int*   labels = (const int*)d_in[1];

    float* ws   = (float*)d_ws;
    float* sq   = ws;                 // 4096 f32
    float* d_ap = ws + NROWS;         // 4096 f32
    float* d_an = ws + 2 * NROWS;     // 4096 f32
    float* d_pn = ws + 3 * NROWS;     // 4096 f32
    int*   ip   = (int*)(ws + 4 * NROWS);
    int*   in_  = (int*)(ws + 5 * NROWS);

    k_rowsumsq<<<NROWS / 8, 256, 0, stream>>>(X, sq);
    k_hardmine<<<NROWS / 16, 256, 0, stream>>>(X, labels, sq, d_ap, d_an, ip, in_);
    k_dpn<<<NROWS / 8, 256, 0, stream>>>(X, sq, ip, in_, d_pn);
    k_loss<<<1, 256, 0, stream>>>(d_ap, d_an, d_pn, (float*)d_out);
}